// RNN_69088843923476
// MI455X (gfx1250) — compile-verified
//
#include <hip/hip_runtime.h>

typedef __attribute__((ext_vector_type(2))) float v2f;
typedef __attribute__((ext_vector_type(8))) float v8f;

#define DHN    512
#define TSTEPS 2048
#define NBATCH 16
#define TS     16   // xs staging depth (LDS tile of 512 x 16 floats)

// ---------------------------------------------------------------------------
// Kernel A: wT[b] = transpose( (|_w| * m_diag[col]) @ nm_signal[b] )
// 16 batches x (32x32) 16x16 tiles, one wave per tile, K in steps of 4 via
// V_WMMA_F32_16X16X4_F32 (exact fp32 matrix path).
// ---------------------------------------------------------------------------
__global__ void __launch_bounds__(256) wprep_kernel(
    const float* __restrict__ W, const float* __restrict__ md,
    const float* __restrict__ nm, float* __restrict__ wT)
{
  const int lane = threadIdx.x & 31;
  const int gw   = blockIdx.x * (blockDim.x >> 5) + (threadIdx.x >> 5);
  const int b    = gw >> 10;          // 1024 tiles per batch
  const int tile = gw & 1023;
  const int tm   = tile >> 5;         // 0..31 row-tile of w
  const int tn   = tile & 31;         // 0..31 col-tile of w
  const int l16  = lane & 15;
  const int kh   = lane >> 4;         // lane-half selects K offset {0,2}
  const int row  = tm * 16 + l16;     // A row (M)
  const int ncol = tn * 16 + l16;     // B col (N)
  const float* __restrict__ nb = nm + (size_t)b * DHN * DHN;

  v8f c = {};
  for (int k = 0; k < DHN; k += 4) {
    const int ca = k + 2 * kh;
    v2f a, bb;
    // A[row][ca], A[row][ca+1]  (16x4 f32 layout: v0=K0|K2, v1=K1|K3)
    a.x = fabsf(W[row * DHN + ca])     * md[ca];
    a.y = fabsf(W[row * DHN + ca + 1]) * md[ca + 1];
    // B[ca][ncol], B[ca+1][ncol] (4x16, rows striped across lanes)
    bb.x = nb[(size_t)ca * DHN + ncol];
    bb.y = nb[(size_t)(ca + 1) * DHN + ncol];
    c = __builtin_amdgcn_wmma_f32_16x16x4_f32(false, a, false, bb,
                                              (short)0, c, false, false);
  }

  // C/D layout: VGPR i -> M = i + 8*(lane>>4), N = lane&15. Store TRANSPOSED:
  // wT[b][col][row] so the time loop reads coalesced along rows.
  float* __restrict__ ob = wT + (size_t)b * DHN * DHN;
  const int mb = 8 * kh;
#pragma unroll
  for (int i = 0; i < 8; ++i)
    ob[(size_t)(tn * 16 + l16) * DHN + (tm * 16 + mb + i)] = c[i];
}

// ---------------------------------------------------------------------------
// Kernel B: persistent per-batch rollout. 16 blocks x 512 threads (16 waves).
// thread tid owns state row tid. Per step: r -> LDS, coalesced wT matvec,
// state update, wave-shuffle + LDS reduction for the scalar output, and
// LDS-staged transpose writes of xs (time-minor) every TS steps.
// ---------------------------------------------------------------------------
__global__ void __launch_bounds__(512) rnn_kernel(
    const float* __restrict__ wT,   const float* __restrict__ u,
    const float* __restrict__ x0,   const float* __restrict__ win,
    const float* __restrict__ traw, const float* __restrict__ wout,
    const float* __restrict__ bias, const float* __restrict__ noise,
    float* __restrict__ outs, float* __restrict__ xs)
{
  __shared__ __align__(16) float rbuf[DHN];
  __shared__ float xtile[DHN * (TS + 1)];   // stride 17: bank-conflict-free
  __shared__ float wred[16];

  const int tid  = threadIdx.x;
  const int b    = blockIdx.x;
  const int lane = tid & 31;
  const int wid  = tid >> 5;

  const float* __restrict__ wTb = wT + (size_t)b * DHN * DHN;
  float* __restrict__ xsb = xs + (size_t)b * DHN * TSTEPS;

  float x     = x0[b * DHN + tid];
  float sg    = 1.f / (1.f + __expf(-traw[tid]));
  float alpha = 1.f / (4.f + 16.f * sg);        // DT / (sig*(20-4)+4)
  float oma   = 1.f - alpha;
  float winv  = win[tid];
  float woutv = wout[tid];
  float biasv = bias[0];

  for (int t = 0; t < TSTEPS; ++t) {
    rbuf[tid] = 1.f / (1.f + __expf(-x));
    __syncthreads();

    // row-tid matvec: acc = sum_k wT[k][tid] * r[k]; coalesced global reads,
    // r chunk broadcast from LDS as float4.
    float acc = 0.f;
    const float4* __restrict__ r4 = (const float4*)rbuf;
#pragma unroll 4
    for (int k = 0; k < DHN; k += 4) {
      float4 rv = r4[k >> 2];
      acc = fmaf(wTb[(size_t)(k + 0) * DHN + tid], rv.x, acc);
      acc = fmaf(wTb[(size_t)(k + 1) * DHN + tid], rv.y, acc);
      acc = fmaf(wTb[(size_t)(k + 2) * DHN + tid], rv.z, acc);
      acc = fmaf(wTb[(size_t)(k + 3) * DHN + tid], rv.w, acc);
    }

    float ut = u[b * TSTEPS + t];
    float nz = noise[((size_t)t * NBATCH + b) * DHN + tid];
    float xn = oma * x + alpha * acc + winv * ut + nz * 0.1f;
    x = xn;
    xtile[tid * (TS + 1) + (t & (TS - 1))] = xn;

    // out_t = wout . sigmoid(x_new) + bias : wave32 shuffle reduce + LDS
    float v = woutv * (1.f / (1.f + __expf(-xn)));
#pragma unroll
    for (int off = 16; off >= 1; off >>= 1) v += __shfl_down(v, off, 32);
    if (lane == 0) wred[wid] = v;
    __syncthreads();
    if (tid == 0) {
      float s = 0.f;
#pragma unroll
      for (int i = 0; i < 16; ++i) s += wred[i];
      outs[b * TSTEPS + t] = s + biasv;
    }

    // flush staged xs tile as coalesced float4 stores every TS steps
    if ((t & (TS - 1)) == (TS - 1)) {
      const int t0 = t - (TS - 1);
#pragma unroll
      for (int cc = 0; cc < 4; ++cc) {          // 2048 float4 chunks / 512 thr
        int q  = cc * DHN + tid;
        int d  = q >> 2;                        // 4 float4-chunks per row
        int s4 = (q & 3) * 4;
        float4 vv;
        vv.x = xtile[d * (TS + 1) + s4 + 0];
        vv.y = xtile[d * (TS + 1) + s4 + 1];
        vv.z = xtile[d * (TS + 1) + s4 + 2];
        vv.w = xtile[d * (TS + 1) + s4 + 3];
        *(float4*)(xsb + (size_t)d * TSTEPS + t0 + s4) = vv;
      }
      __syncthreads();
    }
  }
}

extern "C" void kernel_launch(void* const* d_in, const int* in_sizes, int n_in,
                              void* d_out, int out_size, void* d_ws, size_t ws_size,
                              hipStream_t stream) {
  const float* u     = (const float*)d_in[0];
  const float* x0    = (const float*)d_in[1];
  const float* nm    = (const float*)d_in[2];
  const float* win   = (const float*)d_in[3];
  const float* md    = (const float*)d_in[4];
  const float* wout  = (const float*)d_in[5];
  const float* W     = (const float*)d_in[6];
  const float* traw  = (const float*)d_in[7];
  const float* bias  = (const float*)d_in[8];
  const float* noise = (const float*)d_in[9];

  float* out = (float*)d_out;                 // outputs (B,T,1) first
  float* xs  = out + NBATCH * TSTEPS;         // then xs (B,DH,T)
  float* wT  = (float*)d_ws;                  // 16*512*512*4 = 16 MiB scratch

  // 16 batches * 1024 tiles = 16384 waves; 8 waves/block -> 2048 blocks
  wprep_kernel<<<2048, 256, 0, stream>>>(W, md, nm, wT);
  rnn_kernel<<<NBATCH, DHN, 0, stream>>>(wT, u, x0, win, traw, wout, bias,
                                         noise, out, xs);
}